// MyCell_91104846283147
// MI455X (gfx1250) — compile-verified
//
#include <hip/hip_runtime.h>

// CSR SpMM (sum aggregation): out[row] = sum of x[col[j]] for j in [rowptr[row], rowptr[row+1]).
// Bandwidth-bound gather-reduce (~850MB moved, ~36us floor at 23.3TB/s; x is
// 25.6MB so the random gather is served out of the 192MB L2 after first touch).
// Strategy: one wave32 per row, lane owns 4 floats (b128 per lane -> 512B
// coalesced per edge), gathered rows staged into LDS via the CDNA5 async
// global->LDS engine (ASYNCcnt), double-buffered, NT hints on streaming
// col/out so the x table stays resident in L2.

#define D_FEAT 128
#define WAVES_PER_BLOCK 8
#define BLOCK_THREADS (WAVES_PER_BLOCK * 32)
#define BATCH 4 // edges staged per async batch (4 x 512B in flight per phase)

typedef float v4f __attribute__((ext_vector_type(4)));
// Must match the builtin's canonical V4i type (GCC vector_size style).
typedef int v4i_b __attribute__((vector_size(16)));

#if defined(__gfx1250__) && __has_builtin(__builtin_amdgcn_global_load_async_to_lds_b128)
#define USE_ASYNC_GATHER 1
#else
#define USE_ASYNC_GATHER 0
#endif

#if USE_ASYNC_GATHER
#if __has_builtin(__builtin_amdgcn_s_wait_asynccnt)
#define WAIT_ASYNCCNT(n) __builtin_amdgcn_s_wait_asynccnt(n)
#else
#define WAIT_ASYNCCNT(n) asm volatile("s_wait_asynccnt %0" ::"i"(n) : "memory")
#endif
#endif

__global__ __launch_bounds__(BLOCK_THREADS) void spmm_csr_sum_kernel(
    const float* __restrict__ x, const int* __restrict__ rowptr,
    const int* __restrict__ col, float* __restrict__ out, int n_nodes) {
  const int lane = threadIdx.x & 31;
  const int wave = threadIdx.x >> 5;
  const int row = blockIdx.x * WAVES_PER_BLOCK + wave;

#if USE_ASYNC_GATHER
  // 2 phases x BATCH rows x 512B per wave = 4KB/wave, 32KB/block (of 320KB/WGP).
  __shared__ __align__(16) float smem[WAVES_PER_BLOCK * 2 * BATCH * D_FEAT];
#endif

  if (row >= n_nodes) return;

  const int start = rowptr[row];
  const int end = rowptr[row + 1];

  v4f acc = {0.0f, 0.0f, 0.0f, 0.0f};

#if USE_ASYNC_GATHER
  float* mybuf = smem + wave * (2 * BATCH * D_FEAT) + lane * 4;

  if (end > start) {
    const int nb = (end - start + BATCH - 1) / BATCH;

    auto issue = [&](int b, int phase) {
      const int base = start + b * BATCH;
#pragma unroll
      for (int k = 0; k < BATCH; ++k) {
        int j = base + k;
        int jc = (j < end) ? j : (end - 1); // clamp: keeps in-flight count fixed
        int c = __builtin_nontemporal_load(col + jc); // streaming, keep L2 for x
        const float* src = x + (size_t)c * D_FEAT + lane * 4;
        float* dst = mybuf + phase * (BATCH * D_FEAT) + k * D_FEAT;
        // One instruction moves a whole 512B gathered row (32 lanes x b128).
        __builtin_amdgcn_global_load_async_to_lds_b128(
            (__attribute__((address_space(1))) v4i_b*)src,
            (__attribute__((address_space(3))) v4i_b*)dst, 0, 0);
      }
    };

    auto consume = [&](int b, int phase) {
      const int base = start + b * BATCH;
#pragma unroll
      for (int k = 0; k < BATCH; ++k) {
        if (base + k < end) {
          v4f v = *(const v4f*)(mybuf + phase * (BATCH * D_FEAT) + k * D_FEAT);
          acc += v;
        }
      }
    };

    issue(0, 0);
    int phase = 0;
    for (int b = 0; b < nb; ++b) {
      if (b + 1 < nb) {
        issue(b + 1, phase ^ 1);      // keep next batch in flight
        WAIT_ASYNCCNT(BATCH);         // batch b complete (in-order retirement)
      } else {
        WAIT_ASYNCCNT(0);             // tail: drain
      }
      asm volatile("" ::: "memory");  // don't hoist LDS reads above the wait
      consume(b, phase);
      phase ^= 1;
    }
  }
#else
  // Fallback: direct b128 gathers to VGPRs, 4-deep unroll for MLP.
  int j = start;
  for (; j + (BATCH - 1) < end; j += BATCH) {
    v4f v[BATCH];
#pragma unroll
    for (int k = 0; k < BATCH; ++k) {
      int c = __builtin_nontemporal_load(col + j + k);
      v[k] = *(const v4f*)(x + (size_t)c * D_FEAT + lane * 4);
    }
#pragma unroll
    for (int k = 0; k < BATCH; ++k) acc += v[k];
  }
  for (; j < end; ++j) {
    int c = __builtin_nontemporal_load(col + j);
    acc += *(const v4f*)(x + (size_t)c * D_FEAT + lane * 4);
  }
#endif

  // Streaming write-once output: non-temporal store, keep L2 hot for x.
  v4f* o = (v4f*)(out + (size_t)row * D_FEAT);
  __builtin_nontemporal_store(acc, o + lane);
}

extern "C" void kernel_launch(void* const* d_in, const int* in_sizes, int n_in,
                              void* d_out, int out_size, void* d_ws, size_t ws_size,
                              hipStream_t stream) {
  (void)n_in;
  (void)out_size;
  (void)d_ws;
  (void)ws_size;
  const float* x = (const float*)d_in[0];
  const int* rowptr = (const int*)d_in[1];
  const int* col = (const int*)d_in[2];
  float* out = (float*)d_out;

  const int n_nodes = in_sizes[1] - 1; // rowptr has n_nodes + 1 entries
  const int blocks = (n_nodes + WAVES_PER_BLOCK - 1) / WAVES_PER_BLOCK;
  spmm_csr_sum_kernel<<<blocks, BLOCK_THREADS, 0, stream>>>(x, rowptr, col, out,
                                                            n_nodes);
}